// BoundaryKDV3_24979529793875
// MI455X (gfx1250) — compile-verified
//
#include <hip/hip_runtime.h>
#include <hip/hip_bf16.h>

typedef __attribute__((ext_vector_type(2))) float v2f;
typedef __attribute__((ext_vector_type(8))) float v8f;

#define NCLS 14
#define DIMN 96
#define VOX (96*96*96)
#define VPT 8          // voxels per thread
#define BLK 256        // 8 waves of 32

__global__ void bkd_init_ws(float* ws) {
    int i = threadIdx.x;
    if (i < 64) ws[i] = 0.0f;
}

// ws layout: [0..31]  sums  (b*16 + k)
//            [32..63] counts(b*16 + k)
__global__ __launch_bounds__(BLK) void bkd_main(const float* __restrict__ predS,
                                                const float* __restrict__ predT,
                                                const int*   __restrict__ labels,
                                                float* __restrict__ ws) {
    const int b = blockIdx.y;
    const float* S = predS + (size_t)b * NCLS * VOX;
    const float* T = predT + (size_t)b * NCLS * VOX;
    const int*   L = labels + (size_t)b * VOX;

    float acc[NCLS];
    float cnt[NCLS];
#pragma unroll
    for (int k = 0; k < NCLS; ++k) { acc[k] = 0.0f; cnt[k] = 0.0f; }

    const int base = blockIdx.x * (BLK * VPT) + threadIdx.x;

    for (int i = 0; i < VPT; ++i) {
        const int idx = base + i * BLK;
        if (idx >= VOX) break;
        const int d  = idx % DIMN;
        const int t1 = idx / DIMN;
        const int w  = t1 % DIMN;
        const int h  = t1 / DIMN;

        // ---- per-voxel KL(T||S) over classes (channel-strided, NT-hinted loads)
        float sv[NCLS], tv[NCLS];
        float mS = -3.402823466e38f, mT = -3.402823466e38f;
#pragma unroll
        for (int c = 0; c < NCLS; ++c) {
            sv[c] = __builtin_nontemporal_load(S + (size_t)c * VOX + idx);
            tv[c] = __builtin_nontemporal_load(T + (size_t)c * VOX + idx);
            mS = fmaxf(mS, sv[c]);
            mT = fmaxf(mT, tv[c]);
        }
        float sumS = 0.0f;
#pragma unroll
        for (int c = 0; c < NCLS; ++c) sumS += __expf(sv[c] - mS);
        const float lseS = mS + __logf(sumS);

        float sumT = 0.0f, dot = 0.0f;
#pragma unroll
        for (int c = 0; c < NCLS; ++c) {
            const float e = __expf(tv[c] - mT);
            sumT += e;
            dot  += e * (tv[c] - sv[c]);
        }
        // kl = sum_c pT*(logT - logS) = dot/sumT - lseT + lseS
        const float kl = dot / sumT - (mT + __logf(sumT)) + lseS;

        // ---- boundary mask: class-k boundary iff neighbor-count in (0,26)
        unsigned present = 0u;
        int  first = 0;
        bool haveFirst = false;
        bool same = true;
        const bool interior = (h > 0) & (h < DIMN - 1) &
                              (w > 0) & (w < DIMN - 1) &
                              (d > 0) & (d < DIMN - 1);
#pragma unroll
        for (int dh = -1; dh <= 1; ++dh)
#pragma unroll
        for (int dw = -1; dw <= 1; ++dw)
#pragma unroll
        for (int dd = -1; dd <= 1; ++dd) {
            if (dh == 0 && dw == 0 && dd == 0) continue;
            const int hh = h + dh, ww = w + dw, zz = d + dd;
            const bool inb = ((unsigned)hh < (unsigned)DIMN) &
                             ((unsigned)ww < (unsigned)DIMN) &
                             ((unsigned)zz < (unsigned)DIMN);
            const int ch = hh < 0 ? 0 : (hh > DIMN - 1 ? DIMN - 1 : hh);
            const int cw = ww < 0 ? 0 : (ww > DIMN - 1 ? DIMN - 1 : ww);
            const int cz = zz < 0 ? 0 : (zz > DIMN - 1 ? DIMN - 1 : zz);
            const int val = L[(ch * DIMN + cw) * DIMN + cz];
            if (!haveFirst) { first = val; haveFirst = true; }
            same = same && (val == first);
            present |= inb ? (1u << val) : 0u;
        }
        if (interior && same) present &= ~(1u << first);

        // ---- branchless masked accumulation into per-lane registers
#pragma unroll
        for (int k = 0; k < NCLS; ++k) {
            const float m = (float)((present >> k) & 1u);
            acc[k] = fmaf(m, kl, acc[k]);
            cnt[k] += m;
        }
    }

    // ---- per-wave exact-f32 reduction via V_WMMA_F32_16X16X4_F32
    // Stage each wave's 32(lanes) x 16(classes) partials (+counts) in LDS,
    // reduce columns with A=ones(16x4) chained over 8 K-slices.
    __shared__ float red[BLK / 32][32][32];   // [wave][lane][0:16 sums | 16:32 counts]

    const int wv   = threadIdx.x >> 5;
    const int lane = threadIdx.x & 31;
    float* row = &red[wv][lane][0];
#pragma unroll
    for (int k = 0; k < NCLS; ++k) { row[k] = acc[k]; row[16 + k] = cnt[k]; }
    row[14] = 0.0f; row[15] = 0.0f; row[30] = 0.0f; row[31] = 0.0f;
    __syncthreads();

    // A (16x4 f32): all ones -> every D row = column sum of B
    v2f a; a[0] = 1.0f; a[1] = 1.0f;
    v8f cs = {};   // class sums
    v8f cc = {};   // class counts
    const int r0  = (lane & 16) ? 2 : 0;   // B rows K=0/1 (lanes 0-15), K=2/3 (lanes 16-31)
    const int col = lane & 15;             // B column N
#pragma unroll
    for (int t = 0; t < 8; ++t) {
        v2f bs, bc;
        bs[0] = red[wv][4 * t + r0    ][col];
        bs[1] = red[wv][4 * t + r0 + 1][col];
        bc[0] = red[wv][4 * t + r0    ][16 + col];
        bc[1] = red[wv][4 * t + r0 + 1][16 + col];
        cs = __builtin_amdgcn_wmma_f32_16x16x4_f32(false, a, false, bs, (short)0, cs, false, false);
        cc = __builtin_amdgcn_wmma_f32_16x16x4_f32(false, a, false, bc, (short)0, cc, false, false);
    }
    // D VGPR0: lanes 0-15 hold row M=0, N=lane -> the column sums
    if (lane < NCLS) {
        atomicAdd(&ws[b * 16 + lane],       cs[0]);
        atomicAdd(&ws[32 + b * 16 + lane],  cc[0]);
    }
}

__global__ void bkd_finalize(const float* __restrict__ ws, float* __restrict__ out) {
    if (threadIdx.x == 0 && blockIdx.x == 0) {
        float total = 0.0f, valid = 0.0f;
        for (int b = 0; b < 2; ++b) {
            for (int k = 0; k < NCLS; ++k) {
                const float n = ws[32 + b * 16 + k];
                const float s = ws[b * 16 + k];
                if (n > 0.0f) {
                    total += s / ((float)NCLS * n);
                    valid += 1.0f;
                }
            }
        }
        out[0] = (valid > 0.0f) ? (total / valid) : 0.0f;
    }
}

extern "C" void kernel_launch(void* const* d_in, const int* in_sizes, int n_in,
                              void* d_out, int out_size, void* d_ws, size_t ws_size,
                              hipStream_t stream) {
    const float* predS  = (const float*)d_in[0];
    const float* predT  = (const float*)d_in[1];
    const int*   labels = (const int*)d_in[2];
    float* out = (float*)d_out;
    float* ws  = (float*)d_ws;

    bkd_init_ws<<<1, 64, 0, stream>>>(ws);

    dim3 grid((VOX + BLK * VPT - 1) / (BLK * VPT), 2);  // (432, 2)
    bkd_main<<<grid, BLK, 0, stream>>>(predS, predT, labels, ws);

    bkd_finalize<<<1, 32, 0, stream>>>(ws, out);
}